// SpatialNonIntersectionAxiom_17583596110441
// MI455X (gfx1250) — compile-verified
//
#include <hip/hip_runtime.h>
#include <math.h>

#define E_EDGES 4096
#define N_NODES 512
#define EPS_V   0.001f
#define PROX_V  0.15f
#define TILES_PER_DIM (E_EDGES / 16)          // 256
#define TOTAL_TILES   (TILES_PER_DIM * TILES_PER_DIM)  // 65536
#define WAVES_PER_BLOCK 8

typedef float v2f __attribute__((ext_vector_type(2)));
typedef float v8f __attribute__((ext_vector_type(8)));

__device__ __forceinline__ float clamp01(float x) {
    return fminf(fmaxf(x, 0.0f), 1.0f);
}

// ---------------------------------------------------------------------------
// Pass 1: per-edge precompute (4096 edges) + zero the loss accumulators.
// ws float layout (each E_EDGES long): psx psy dx dy sq dda mx my m2 half
// ws int layout  (each E_EDGES long): batch src dst
// ---------------------------------------------------------------------------
__global__ void edge_pre_kernel(const float* __restrict__ pos,
                                const int* __restrict__ eidx,
                                float* __restrict__ wf,
                                int* __restrict__ wi,
                                float* accSum, unsigned* accCnt) {
    int e = blockIdx.x * blockDim.x + threadIdx.x;
    if (e == 0) { *accSum = 0.0f; *accCnt = 0u; }
    if (e >= E_EDGES) return;

    int s = eidx[e];
    int d = eidx[E_EDGES + e];
    float px = pos[2 * s + 0], py = pos[2 * s + 1];
    float qx = pos[2 * d + 0], qy = pos[2 * d + 1];
    float dx = qx - px, dy = qy - py;
    float n2 = dx * dx + dy * dy;
    float mx = 0.5f * (px + qx), my = 0.5f * (py + qy);

    wf[0 * E_EDGES + e] = px;
    wf[1 * E_EDGES + e] = py;
    wf[2 * E_EDGES + e] = dx;
    wf[3 * E_EDGES + e] = dy;
    wf[4 * E_EDGES + e] = fmaxf(n2, 1e-12f);          // sq (clamped)
    wf[5 * E_EDGES + e] = dx * px + dy * py;          // d1 . p_src
    wf[6 * E_EDGES + e] = mx;
    wf[7 * E_EDGES + e] = my;
    wf[8 * E_EDGES + e] = mx * mx + my * my;          // |mid|^2
    wf[9 * E_EDGES + e] = 0.5f * sqrtf(n2);           // half (unclamped norm)
    wi[0 * E_EDGES + e] = s / N_NODES;                // batch
    wi[1 * E_EDGES + e] = s;
    wi[2 * E_EDGES + e] = d;
}

// ---------------------------------------------------------------------------
// Pass 2: one wave32 per 16x16 tile of the E x E pair field.
// 4x V_WMMA_F32_16X16X4_F32 build the Gram matrices d.d', d.p', p.d', m.m'
// (K padded 2->4 with zeros); VALU does clip/sqrt/mask; NT b32 stores stream
// the two 64 MB outputs.
// ---------------------------------------------------------------------------
__global__ __launch_bounds__(256) void
pair_tile_kernel(const float* __restrict__ wf, const int* __restrict__ wi,
                 float* __restrict__ mask_out, float* __restrict__ score_out,
                 float* accSum, unsigned* accCnt) {
    const float* psx = wf + 0 * E_EDGES;
    const float* psy = wf + 1 * E_EDGES;
    const float* dxa = wf + 2 * E_EDGES;
    const float* dya = wf + 3 * E_EDGES;
    const float* sqa = wf + 4 * E_EDGES;
    const float* dda = wf + 5 * E_EDGES;
    const float* mxa = wf + 6 * E_EDGES;
    const float* mya = wf + 7 * E_EDGES;
    const float* m2a = wf + 8 * E_EDGES;
    const float* hfa = wf + 9 * E_EDGES;
    const int*  bata = wi + 0 * E_EDGES;
    const int*  srca = wi + 1 * E_EDGES;
    const int*  dsta = wi + 2 * E_EDGES;

    const int lane = threadIdx.x & 31;
    const int wave = threadIdx.x >> 5;
    const int tile = blockIdx.x * WAVES_PER_BLOCK + wave;
    const int rb = (tile >> 8) << 4;     // tile row base
    const int cb = (tile & 255) << 4;    // tile col base
    const int l15 = lane & 15;
    const bool hi = lane >= 16;          // upper half-wave carries K=2,3 -> zero pad

    // A (rows): 16x4 f32, K = (val.x, val.y, 0, 0)
    const int mA = rb + l15;
    const int nB = cb + l15;
    v2f aD, aP, aM, bD, bP, bM;
    aD.x = hi ? 0.0f : dxa[mA];  aD.y = hi ? 0.0f : dya[mA];
    aP.x = hi ? 0.0f : psx[mA];  aP.y = hi ? 0.0f : psy[mA];
    aM.x = hi ? 0.0f : mxa[mA];  aM.y = hi ? 0.0f : mya[mA];
    // B (cols): 4x16 f32, rows K=0 / K=1 carry the 2-vector components
    bD.x = hi ? 0.0f : dxa[nB];  bD.y = hi ? 0.0f : dya[nB];
    bP.x = hi ? 0.0f : psx[nB];  bP.y = hi ? 0.0f : psy[nB];
    bM.x = hi ? 0.0f : mxa[nB];  bM.y = hi ? 0.0f : mya[nB];

    v8f z = {0.f, 0.f, 0.f, 0.f, 0.f, 0.f, 0.f, 0.f};
    v8f Gdd = __builtin_amdgcn_wmma_f32_16x16x4_f32(false, aD, false, bD, (short)0, z, false, false);
    v8f Gdp = __builtin_amdgcn_wmma_f32_16x16x4_f32(false, aD, false, bP, (short)0, z, false, false);
    v8f Gpd = __builtin_amdgcn_wmma_f32_16x16x4_f32(false, aP, false, bD, (short)0, z, false, false);
    v8f Gmm = __builtin_amdgcn_wmma_f32_16x16x4_f32(false, aM, false, bM, (short)0, z, false, false);

    // Column-side scalars for this lane (C/D layout: N = lane & 15).
    const int n = cb + l15;
    const float e_sq = sqa[n], dda_c = dda[n], m2_c = m2a[n], hf_c = hfa[n];
    const float pcx = psx[n], pcy = psy[n], dcx = dxa[n], dcy = dya[n];
    const int bat_c = bata[n], s_c = srca[n], d_c = dsta[n];

    float sum = 0.0f;
    unsigned cnt = 0u;
    const int rowOfs = hi ? 8 : 0;       // C/D layout: lanes 16-31 hold M = r + 8

    #pragma unroll
    for (int r = 0; r < 8; ++r) {
        const int m = rb + r + rowOfs;
        const float a_sq = sqa[m], dda_r = dda[m], m2_r = m2a[m], hf_r = hfa[m];
        const float prx = psx[m], pry = psy[m], drx = dxa[m], dry = dya[m];
        const int bat_r = bata[m], s_r = srca[m], d_r = dsta[m];

        const float b = Gdd[r];
        const float c = dda_r - Gdp[r];          // d_i.p_i - d_i.p_j
        const float f = Gpd[r] - dda_c;          // p_i.d_j - d_j.p_j
        const float denom = fmaxf(a_sq * e_sq - b * b, 1e-12f);
        float s = (b * f - c * e_sq) / denom;
        float t = (a_sq * f - b * c) / denom;
        s = clamp01(s);
        t = clamp01(t);
        t = clamp01((b * s + f) / e_sq);
        s = clamp01((b * t - c) / a_sq);
        const float ddx = (prx + s * drx) - (pcx + t * dcx);
        const float ddy = (pry + s * dry) - (pcy + t * dcy);
        const float dmin = sqrtf(ddx * ddx + ddy * ddy);

        const float mid2 = fmaxf(m2_r + m2_c - 2.0f * Gmm[r], 0.0f);
        const float dist = sqrtf(mid2);
        const bool shares = (s_r == s_c) | (s_r == d_c) | (d_r == s_c) | (d_r == d_c);
        const bool mk = (dist < hf_r + hf_c + PROX_V) && (n > m) && (bat_r == bat_c) && !shares;

        const float pl = mk ? fmaxf(EPS_V - dmin, 0.0f) : 0.0f;
        const size_t off = (size_t)m * E_EDGES + (size_t)n;
        __builtin_nontemporal_store((mk && pl > 0.0f) ? 1.0f : 0.0f, mask_out + off);
        __builtin_nontemporal_store(pl, score_out + off);
        sum += pl;
        cnt += mk ? 1u : 0u;
    }

    // wave32 reduction, one atomic per wave
    #pragma unroll
    for (int o = 16; o >= 1; o >>= 1) {
        sum += __shfl_xor(sum, o, 32);
        cnt += __shfl_xor(cnt, o, 32);
    }
    if (lane == 0) {
        atomicAdd(accSum, sum);
        atomicAdd(accCnt, cnt);
    }
}

// ---------------------------------------------------------------------------
// Pass 3: finalize loss = sum / max(n_pairs, 1)
// ---------------------------------------------------------------------------
__global__ void finalize_kernel(const float* accSum, const unsigned* accCnt,
                                float* __restrict__ out) {
    if (threadIdx.x == 0 && blockIdx.x == 0) {
        const float s = *accSum;
        const float c = (float)(*accCnt);
        out[0] = s / fmaxf(c, 1.0f);
    }
}

extern "C" void kernel_launch(void* const* d_in, const int* in_sizes, int n_in,
                              void* d_out, int out_size, void* d_ws, size_t ws_size,
                              hipStream_t stream) {
    const float* pos  = (const float*)d_in[0];   // (8,512,2) flattened
    // d_in[1] = adjacency: unused by the reference
    const int*   eidx = (const int*)d_in[2];     // (2,4096)

    float* out = (float*)d_out;                  // [loss | mask E*E | scores E*E]
    float* mask_out  = out + 1;
    float* score_out = out + 1 + (size_t)E_EDGES * E_EDGES;

    float* wf = (float*)d_ws;                    // 10 * 4096 floats
    int*   wi = (int*)(wf + 10 * E_EDGES);       // 3 * 4096 ints
    float* accSum = (float*)(wi + 3 * E_EDGES);
    unsigned* accCnt = (unsigned*)(accSum + 1);

    edge_pre_kernel<<<(E_EDGES + 255) / 256, 256, 0, stream>>>(pos, eidx, wf, wi, accSum, accCnt);
    pair_tile_kernel<<<TOTAL_TILES / WAVES_PER_BLOCK, 256, 0, stream>>>(
        wf, wi, mask_out, score_out, accSum, accCnt);
    finalize_kernel<<<1, 1, 0, stream>>>(accSum, accCnt, out);
}